// verletNetworks_58755152609566
// MI455X (gfx1250) — compile-verified
//
#include <hip/hip_runtime.h>
#include <hip/hip_bf16.h>

// ---------------------------------------------------------------------------
// CDNA5 / gfx1250 implementation of the verlet graph network reference.
// All dense math runs on V_WMMA_F32_16X16X4_F32 (fp32 tensor path, matches
// reference precision). wave32: each wave owns one 16-column tile and the
// full 64-row output (4 M-tiles of 16).
// ---------------------------------------------------------------------------

typedef float v2f_t __attribute__((ext_vector_type(2)));
typedef float v8f_t __attribute__((ext_vector_type(8)));

#define WMMA_F32X4(a, b, c)                                                    \
  __builtin_amdgcn_wmma_f32_16x16x4_f32(false, (a), false, (b), (short)0, (c), \
                                        false, false)

static constexpr float HSTEP = 0.1f;
static constexpr float EPS_TV = 1e-3f;

// ---------------------------------------------------------------------------
__global__ void fill_zero(float* __restrict__ p, int n) {
  for (int i = blockIdx.x * blockDim.x + threadIdx.x; i < n;
       i += gridDim.x * blockDim.x)
    p[i] = 0.f;
}

// ---------------------------------------------------------------------------
// OutT[i][c] = sum_k W[c][k] * X[k][i];  W: [64][16], X: [16][num],
// OutT: [num][64].   K = 16 -> 4 WMMA k-steps, 4 M-tiles.
__global__ __launch_bounds__(128) void open_kernel(
    const float* __restrict__ W, const float* __restrict__ X,
    float* __restrict__ OutT, int num) {
  __shared__ float sW[64 * 16];
  for (int t = threadIdx.x; t < 64 * 16; t += blockDim.x) sW[t] = W[t];
  __syncthreads();

  const int lane = threadIdx.x & 31;
  const int wave = (blockIdx.x * blockDim.x + threadIdx.x) >> 5;
  const int row = lane & 15;  // M row / column index within tile
  const int h = lane >> 4;    // k-half select
  const int ntiles = num >> 4;
  if (wave >= ntiles) return;  // wave-uniform
  const int col = (wave << 4) + row;

  v8f_t acc[4] = {};
#pragma unroll
  for (int s = 0; s < 4; ++s) {
    const int k = 4 * s + 2 * h;
    v2f_t b;
    b.x = X[(size_t)k * num + col];
    b.y = X[(size_t)(k + 1) * num + col];
#pragma unroll
    for (int m = 0; m < 4; ++m) {
      const v2f_t a = *(const v2f_t*)&sW[(m * 16 + row) * 16 + k];
      acc[m] = WMMA_F32X4(a, b, acc[m]);
    }
  }
  // lane(row,h), vgpr r -> channel m*16 + 8*h + r (8 consecutive floats)
  float* dst = OutT + (size_t)col * 64;
#pragma unroll
  for (int m = 0; m < 4; ++m) {
    const int cb = m * 16 + 8 * h;
#pragma unroll
    for (int r = 0; r < 8; ++r) dst[cb + r] = acc[m][r];
  }
}

// ---------------------------------------------------------------------------
// Pass A over edges: Y = Kn @ (xn[:,i]-xn[:,j]); accumulate per-channel
// sum / sumsq of Y over all edges (for tv_norm).  Y is not stored.
__global__ __launch_bounds__(128) void edge_gemm_stats(
    const float* __restrict__ Kn, const float* __restrict__ xnT,
    const int* __restrict__ iInd, const int* __restrict__ jInd,
    float* __restrict__ statS, float* __restrict__ statQ, int E,
    int tilesPerWave) {
  __shared__ float sK[64 * 64];
  for (int t = threadIdx.x; t < 64 * 64; t += blockDim.x) sK[t] = Kn[t];
  __syncthreads();

  const int lane = threadIdx.x & 31;
  const int wave = (blockIdx.x * blockDim.x + threadIdx.x) >> 5;
  const int row = lane & 15;
  const int h = lane >> 4;
  const int ntiles = E >> 4;
  const int tile0 = wave * tilesPerWave;

  v8f_t S[4] = {}, Q[4] = {};
  for (int t = 0; t < tilesPerWave; ++t) {
    const int tile = tile0 + t;
    if (tile >= ntiles) break;  // wave-uniform
    const int e = (tile << 4) + row;
    const float* pI = xnT + (size_t)iInd[e] * 64;
    const float* pJ = xnT + (size_t)jInd[e] * 64;
    v8f_t acc[4] = {};
#pragma unroll
    for (int s = 0; s < 16; ++s) {
      const int k = 4 * s + 2 * h;
      const v2f_t bi = *(const v2f_t*)(pI + k);
      const v2f_t bj = *(const v2f_t*)(pJ + k);
      v2f_t b;
      b.x = bi.x - bj.x;
      b.y = bi.y - bj.y;
#pragma unroll
      for (int m = 0; m < 4; ++m) {
        const v2f_t a = *(const v2f_t*)&sK[(m * 16 + row) * 64 + k];
        acc[m] = WMMA_F32X4(a, b, acc[m]);
      }
    }
#pragma unroll
    for (int m = 0; m < 4; ++m) {
      S[m] += acc[m];
      Q[m] += acc[m] * acc[m];
    }
  }
  // reduce across the 16 lanes of each half, then a few atomics
#pragma unroll
  for (int m = 0; m < 4; ++m) {
#pragma unroll
    for (int r = 0; r < 8; ++r) {
      float s = S[m][r], q = Q[m][r];
      for (int off = 8; off >= 1; off >>= 1) {
        s += __shfl_xor(s, off, 16);
        q += __shfl_xor(q, off, 16);
      }
      if (row == 0) {
        const int c = m * 16 + 8 * h + r;
        atomicAdd(&statS[c], s);
        atomicAdd(&statQ[c], q);
      }
    }
  }
}

// ---------------------------------------------------------------------------
// mean = S/n ; inv = 1/sqrt(sum((y-mean)^2)+eps) = rsqrt(Q - S^2/n + eps)
__global__ void finalize_stats(const float* __restrict__ S,
                               const float* __restrict__ Q,
                               float* __restrict__ mean,
                               float* __restrict__ inv, float count) {
  const int c = threadIdx.x;  // 64 threads
  const float s = S[c], q = Q[c];
  const float m = s / count;
  mean[c] = m;
  inv[c] = rsqrtf(q - s * s / count + EPS_TV);
}

// ---------------------------------------------------------------------------
// Pass B over edges: recompute Y via WMMA, apply tv_norm + relu update to
// xeT, and fuse the edgeDiv scatter:  div[:,i] += xe_new ; div[:,j] -= xe_new.
__global__ __launch_bounds__(128) void edge_update_scatter(
    const float* __restrict__ Kn, const float* __restrict__ xnT,
    const int* __restrict__ iInd, const int* __restrict__ jInd,
    const float* __restrict__ mean, const float* __restrict__ inv,
    float* __restrict__ xeT, float* __restrict__ divT, int E) {
  __shared__ float sK[64 * 64];
  __shared__ float sMean[64], sInv[64];
  for (int t = threadIdx.x; t < 64 * 64; t += blockDim.x) sK[t] = Kn[t];
  if (threadIdx.x < 64) {
    sMean[threadIdx.x] = mean[threadIdx.x];
    sInv[threadIdx.x] = inv[threadIdx.x];
  }
  __syncthreads();

  const int lane = threadIdx.x & 31;
  const int wave = (blockIdx.x * blockDim.x + threadIdx.x) >> 5;
  const int row = lane & 15;
  const int h = lane >> 4;
  const int ntiles = E >> 4;
  if (wave >= ntiles) return;  // wave-uniform

  const int e = (wave << 4) + row;
  const int iN = iInd[e], jN = jInd[e];
  const float* pI = xnT + (size_t)iN * 64;
  const float* pJ = xnT + (size_t)jN * 64;

  v8f_t acc[4] = {};
#pragma unroll
  for (int s = 0; s < 16; ++s) {
    const int k = 4 * s + 2 * h;
    const v2f_t bi = *(const v2f_t*)(pI + k);
    const v2f_t bj = *(const v2f_t*)(pJ + k);
    v2f_t b;
    b.x = bi.x - bj.x;
    b.y = bi.y - bj.y;
#pragma unroll
    for (int m = 0; m < 4; ++m) {
      const v2f_t a = *(const v2f_t*)&sK[(m * 16 + row) * 64 + k];
      acc[m] = WMMA_F32X4(a, b, acc[m]);
    }
  }

  float* xePtr = xeT + (size_t)e * 64;
  float* divI = divT + (size_t)iN * 64;
  float* divJ = divT + (size_t)jN * 64;
#pragma unroll
  for (int m = 0; m < 4; ++m) {
    const int cb = m * 16 + 8 * h;
#pragma unroll
    for (int r = 0; r < 8; ++r) {
      const int c = cb + r;
      const float y = (acc[m][r] - sMean[c]) * sInv[c];
      const float xnew = xePtr[c] + HSTEP * fmaxf(y, 0.f);
      xePtr[c] = xnew;
      atomicAdd(&divI[c], xnew);
      atomicAdd(&divJ[c], -xnew);
    }
  }
}

// ---------------------------------------------------------------------------
// OutT = W @ InT^T (stored transposed) with per-channel sum/sumsq stats.
// W: [64][64], InT/OutT: [num][64].  Used for Ke@div and Kd@xn.
__global__ __launch_bounds__(128) void node_gemm_stats(
    const float* __restrict__ W, const float* __restrict__ InT,
    float* __restrict__ OutT, float* __restrict__ statS,
    float* __restrict__ statQ, int num, int tilesPerWave) {
  __shared__ float sK[64 * 64];
  for (int t = threadIdx.x; t < 64 * 64; t += blockDim.x) sK[t] = W[t];
  __syncthreads();

  const int lane = threadIdx.x & 31;
  const int wave = (blockIdx.x * blockDim.x + threadIdx.x) >> 5;
  const int row = lane & 15;
  const int h = lane >> 4;
  const int ntiles = num >> 4;
  const int tile0 = wave * tilesPerWave;

  v8f_t S[4] = {}, Q[4] = {};
  for (int t = 0; t < tilesPerWave; ++t) {
    const int tile = tile0 + t;
    if (tile >= ntiles) break;  // wave-uniform
    const int col = (tile << 4) + row;
    const float* pIn = InT + (size_t)col * 64;
    v8f_t acc[4] = {};
#pragma unroll
    for (int s = 0; s < 16; ++s) {
      const int k = 4 * s + 2 * h;
      const v2f_t b = *(const v2f_t*)(pIn + k);
#pragma unroll
      for (int m = 0; m < 4; ++m) {
        const v2f_t a = *(const v2f_t*)&sK[(m * 16 + row) * 64 + k];
        acc[m] = WMMA_F32X4(a, b, acc[m]);
      }
    }
    float* dst = OutT + (size_t)col * 64;
#pragma unroll
    for (int m = 0; m < 4; ++m) {
      const int cb = m * 16 + 8 * h;
#pragma unroll
      for (int r = 0; r < 8; ++r) dst[cb + r] = acc[m][r];
      S[m] += acc[m];
      Q[m] += acc[m] * acc[m];
    }
  }
#pragma unroll
  for (int m = 0; m < 4; ++m) {
#pragma unroll
    for (int r = 0; r < 8; ++r) {
      float s = S[m][r], q = Q[m][r];
      for (int off = 8; off >= 1; off >>= 1) {
        s += __shfl_xor(s, off, 16);
        q += __shfl_xor(q, off, 16);
      }
      if (row == 0) {
        const int c = m * 16 + 8 * h + r;
        atomicAdd(&statS[c], s);
        atomicAdd(&statQ[c], q);
      }
    }
  }
}

// ---------------------------------------------------------------------------
// xn -= H * ( relu(tvnorm(U)) + relu(tvnorm(V)) ), elementwise over [N][64].
__global__ void node_update(const float* __restrict__ UT,
                            const float* __restrict__ VT,
                            const float* __restrict__ mU,
                            const float* __restrict__ iU,
                            const float* __restrict__ mV,
                            const float* __restrict__ iV,
                            float* __restrict__ xnT, int total) {
  const int idx = blockIdx.x * blockDim.x + threadIdx.x;
  if (idx >= total) return;
  const int c = idx & 63;
  const float bu = fmaxf((UT[idx] - mU[c]) * iU[c], 0.f);
  const float cv = fmaxf((VT[idx] - mV[c]) * iV[c], 0.f);
  xnT[idx] -= HSTEP * (bu + cv);
}

// ---------------------------------------------------------------------------
// Out[c][i] = sum_k Wc[c][k] * InT[i][k];  Wc: [16][64], Out: [16][num].
__global__ __launch_bounds__(128) void close_kernel(
    const float* __restrict__ Wc, const float* __restrict__ InT,
    float* __restrict__ Out, int num) {
  __shared__ float sW[16 * 64];
  for (int t = threadIdx.x; t < 16 * 64; t += blockDim.x) sW[t] = Wc[t];
  __syncthreads();

  const int lane = threadIdx.x & 31;
  const int wave = (blockIdx.x * blockDim.x + threadIdx.x) >> 5;
  const int row = lane & 15;
  const int h = lane >> 4;
  const int ntiles = num >> 4;
  if (wave >= ntiles) return;  // wave-uniform
  const int col = (wave << 4) + row;
  const float* pIn = InT + (size_t)col * 64;

  v8f_t acc = {};
#pragma unroll
  for (int s = 0; s < 16; ++s) {
    const int k = 4 * s + 2 * h;
    const v2f_t b = *(const v2f_t*)(pIn + k);
    const v2f_t a = *(const v2f_t*)&sW[row * 64 + k];
    acc = WMMA_F32X4(a, b, acc);
  }
#pragma unroll
  for (int r = 0; r < 8; ++r)
    Out[(size_t)(8 * h + r) * num + col] = acc[r];
}

// ---------------------------------------------------------------------------
extern "C" void kernel_launch(void* const* d_in, const int* in_sizes, int n_in,
                              void* d_out, int out_size, void* d_ws,
                              size_t ws_size, hipStream_t stream) {
  const float* xn = (const float*)d_in[0];       // [16][N]
  const float* xe = (const float*)d_in[1];       // [16][E]
  const int* iInd = (const int*)d_in[2];         // [E]
  const int* jInd = (const int*)d_in[3];         // [E]
  const float* KNopen = (const float*)d_in[4];   // [64][16]
  const float* KEopen = (const float*)d_in[5];   // [64][16]
  const float* KNclose = (const float*)d_in[6];  // [16][64]
  const float* KEclose = (const float*)d_in[7];  // [16][64]
  const float* KN = (const float*)d_in[8];       // [4][64][64]
  const float* KE = (const float*)d_in[9];       // [4][64][64]
  const float* KD = (const float*)d_in[10];      // [4][64][64]

  const int N = in_sizes[0] / 16;
  const int E = in_sizes[2];

  float* ws = (float*)d_ws;
  float* xnT = ws;   ws += (size_t)N * 64;   // [N][64]
  float* xeT = ws;   ws += (size_t)E * 64;   // [E][64]
  float* divT = ws;  ws += (size_t)N * 64;   // [N][64]
  float* UT = ws;    ws += (size_t)N * 64;   // [N][64]
  float* VT = ws;    ws += (size_t)N * 64;   // [N][64]
  float* stats = ws;                         // 768 floats
  float *Se = stats, *Qe = stats + 64, *Me = stats + 128, *Ie = stats + 192;
  float *SU = stats + 256, *QU = stats + 320, *MU = stats + 384,
        *IU = stats + 448;
  float *SV = stats + 512, *QV = stats + 576, *MV = stats + 640,
        *IV = stats + 704;

  const int nTilesN = N / 16;
  const int nTilesE = E / 16;
  const dim3 b128(128);
  auto blocksForWaves = [](int waves) { return dim3((waves + 3) / 4); };

  open_kernel<<<blocksForWaves(nTilesN), b128, 0, stream>>>(KNopen, xn, xnT, N);
  open_kernel<<<blocksForWaves(nTilesE), b128, 0, stream>>>(KEopen, xe, xeT, E);

  for (int l = 0; l < 4; ++l) {
    const float* Kn = KN + (size_t)l * 64 * 64;
    const float* Ke = KE + (size_t)l * 64 * 64;
    const float* Kd = KD + (size_t)l * 64 * 64;

    fill_zero<<<4, 256, 0, stream>>>(stats, 768);
    fill_zero<<<2048, 256, 0, stream>>>(divT, N * 64);

    const int wavesE = (nTilesE + 7) / 8;
    edge_gemm_stats<<<blocksForWaves(wavesE), b128, 0, stream>>>(
        Kn, xnT, iInd, jInd, Se, Qe, E, 8);
    finalize_stats<<<1, 64, 0, stream>>>(Se, Qe, Me, Ie, (float)E);
    edge_update_scatter<<<blocksForWaves(nTilesE), b128, 0, stream>>>(
        Kn, xnT, iInd, jInd, Me, Ie, xeT, divT, E);

    const int wavesN = (nTilesN + 3) / 4;
    node_gemm_stats<<<blocksForWaves(wavesN), b128, 0, stream>>>(
        Ke, divT, UT, SU, QU, N, 4);
    node_gemm_stats<<<blocksForWaves(wavesN), b128, 0, stream>>>(
        Kd, xnT, VT, SV, QV, N, 4);
    finalize_stats<<<1, 64, 0, stream>>>(SU, QU, MU, IU, (float)N);
    finalize_stats<<<1, 64, 0, stream>>>(SV, QV, MV, IV, (float)N);
    node_update<<<(N * 64 + 255) / 256, 256, 0, stream>>>(UT, VT, MU, IU, MV,
                                                          IV, xnT, N * 64);
  }

  float* outN = (float*)d_out;              // [16][N]
  float* outE = outN + (size_t)16 * N;      // [16][E]
  close_kernel<<<blocksForWaves(nTilesN), b128, 0, stream>>>(KNclose, xnT, outN,
                                                             N);
  close_kernel<<<blocksForWaves(nTilesE), b128, 0, stream>>>(KEclose, xeT, outE,
                                                             E);
}